// GCN3_16226386444409
// MI455X (gfx1250) — compile-verified
//
#include <hip/hip_runtime.h>

typedef float v2f __attribute__((ext_vector_type(2)));
typedef float v8f __attribute__((ext_vector_type(8)));

#define HID 32
#define BN_EPS 1e-5f

// ---------------- degree / normalization ----------------
__global__ void k_init_deg(float* deg, int N) {
  int i = blockIdx.x * blockDim.x + threadIdx.x;
  if (i < N) deg[i] = 1.0f;                 // self-loop weight
}
__global__ void k_deg_edges(const int* __restrict__ col, const float* __restrict__ w,
                            float* deg, int E) {
  int e = blockIdx.x * blockDim.x + threadIdx.x;
  if (e < E) atomicAdd(&deg[col[e]], w[e]);
}
__global__ void k_deg_rsqrt(float* deg, int N) {
  int i = blockIdx.x * blockDim.x + threadIdx.x;
  if (i < N) deg[i] = rsqrtf(deg[i]);       // deg >= 1 always (self-loops)
}

// ---------------- dense feature transform: D[M x 32] = A[M x K] @ B[K x 32] ----------------
// One wave per 16x16 output tile, V_WMMA_F32_16X16X4_F32, K stepped by 4.
template <int K>
__global__ __launch_bounds__(256) void k_gemm_wmma(const float* __restrict__ A,
                                                   const float* __restrict__ B,
                                                   float* __restrict__ D, int M) {
  const int lane = threadIdx.x;             // 0..31 (wave32)
  const int wave = threadIdx.y;             // 0..7
  const int mt   = blockIdx.x * 8 + wave;   // 16-row tile index
  const int numT = (M + 15) >> 4;
  if (mt >= numT) return;                   // wave-uniform exit: EXEC all-1s inside
  const int nb   = blockIdx.y * 16;         // 0 or 16 (HID == 32)
  const int half = lane >> 4;               // 0: lanes 0-15, 1: lanes 16-31
  const int lm   = lane & 15;

  int arow_i = mt * 16 + lm;
  if (arow_i > M - 1) arow_i = M - 1;       // clamp (keeps EXEC full on tail tile)
  const float* arow = A + (size_t)arow_i * K;

  v8f c = {};
#pragma unroll
  for (int k0 = 0; k0 < K; k0 += 4) {
    const int ka = k0 + 2 * half;
    v2f a;
    a.x = arow[ka];
    a.y = arow[ka + 1];
    v2f b;
    b.x = B[(size_t)ka * HID + nb + lm];
    b.y = B[(size_t)(ka + 1) * HID + nb + lm];
    // 8 args: (neg_a, A, neg_b, B, c_mod, C, reuse_a, reuse_b)
    c = __builtin_amdgcn_wmma_f32_16x16x4_f32(false, a, false, b, (short)0, c,
                                              false, false);
  }
#pragma unroll
  for (int r = 0; r < 8; ++r) {
    const int orow = mt * 16 + r + 8 * half;
    if (orow < M) D[(size_t)orow * HID + nb + lm] = c[r];
  }
}

// ---------------- conv output init with bias ----------------
__global__ void k_init_bias(float* __restrict__ cbuf, const float* __restrict__ b, int total) {
  int i = blockIdx.x * blockDim.x + threadIdx.x;
  if (i < total) cbuf[i] = b[i & (HID - 1)];
}

// ---------------- gather / normalize / scatter-add (one lane per (edge, feature)) ----------------
__global__ void k_prop(const float* __restrict__ h, const int* __restrict__ row,
                       const int* __restrict__ col, const float* __restrict__ w,
                       const float* __restrict__ dis, float* __restrict__ out,
                       int E, int N) {
  long long tid = (long long)blockIdx.x * blockDim.x + threadIdx.x;
  int e = (int)(tid >> 5);
  int f = (int)(tid & 31);
  if (e >= E + N) return;
  int r, c; float ww;
  if (e < E) { r = row[e]; c = col[e]; ww = w[e]; }
  else       { r = e - E;  c = r;      ww = 1.0f; }   // self-loop
  float nrm = dis[r] * ww * dis[c];
  atomicAdd(out + (size_t)c * HID + f, h[(size_t)r * HID + f] * nrm);
}

// ---------------- BN statistics over relu(conv) ----------------
__global__ void k_zero_stats(float* stats) {
  if (threadIdx.x < 64) stats[threadIdx.x] = 0.0f;
}
__global__ void k_bn_stats(const float* __restrict__ c, float* __restrict__ stats, int N) {
  __shared__ float ls[64];
  if (threadIdx.x < 64) ls[threadIdx.x] = 0.0f;
  __syncthreads();
  const int f      = threadIdx.x & 31;
  const int r0     = blockIdx.x * (blockDim.x >> 5) + (threadIdx.x >> 5);
  const int stride = gridDim.x * (blockDim.x >> 5);
  float s = 0.0f, s2 = 0.0f;
  for (int n = r0; n < N; n += stride) {
    float v = fmaxf(c[(size_t)n * HID + f], 0.0f);
    s += v; s2 += v * v;
  }
  atomicAdd(&ls[f], s);
  atomicAdd(&ls[32 + f], s2);
  __syncthreads();
  if (threadIdx.x < 64) atomicAdd(&stats[threadIdx.x], ls[threadIdx.x]);
}

// ---------------- relu + BN apply ----------------
__global__ void k_bn_apply(const float* __restrict__ c, const float* __restrict__ stats,
                           const float* __restrict__ g, const float* __restrict__ be,
                           float* __restrict__ out, int N) {
  int i = blockIdx.x * blockDim.x + threadIdx.x;
  if (i >= N * HID) return;
  const int f = i & 31;
  const float invN = 1.0f / (float)N;
  const float mu   = stats[f] * invN;
  const float var  = stats[32 + f] * invN - mu * mu;   // biased variance
  const float v    = fmaxf(c[i], 0.0f);
  out[i] = (v - mu) * rsqrtf(var + BN_EPS) * g[f] + be[f];
}

// ---------------- final linear: [N x 96] @ [96 x 2] + bl ----------------
__global__ void k_final(const float* __restrict__ o1, const float* __restrict__ o2,
                        const float* __restrict__ o3, const float* __restrict__ Wl,
                        const float* __restrict__ bl, float* __restrict__ out, int N) {
  int n = blockIdx.x * blockDim.x + threadIdx.x;
  if (n >= N) return;
  float a0 = bl[0], a1 = bl[1];
  const float* p = o1 + (size_t)n * HID;
#pragma unroll
  for (int j = 0; j < HID; ++j) { float v = p[j]; a0 += v * Wl[2 * j]; a1 += v * Wl[2 * j + 1]; }
  p = o2 + (size_t)n * HID;
#pragma unroll
  for (int j = 0; j < HID; ++j) { float v = p[j]; a0 += v * Wl[2 * (HID + j)]; a1 += v * Wl[2 * (HID + j) + 1]; }
  p = o3 + (size_t)n * HID;
#pragma unroll
  for (int j = 0; j < HID; ++j) { float v = p[j]; a0 += v * Wl[2 * (2 * HID + j)]; a1 += v * Wl[2 * (2 * HID + j) + 1]; }
  out[2 * n] = a0;
  out[2 * n + 1] = a1;
}

extern "C" void kernel_launch(void* const* d_in, const int* in_sizes, int n_in,
                              void* d_out, int out_size, void* d_ws, size_t ws_size,
                              hipStream_t stream) {
  const float* x   = (const float*)d_in[0];
  const int*   ei  = (const int*)d_in[1];
  const float* ew  = (const float*)d_in[2];
  const float* W1  = (const float*)d_in[3];
  const float* b1  = (const float*)d_in[4];
  const float* g1  = (const float*)d_in[5];
  const float* be1 = (const float*)d_in[6];
  const float* W2  = (const float*)d_in[7];
  const float* b2  = (const float*)d_in[8];
  const float* g2  = (const float*)d_in[9];
  const float* be2 = (const float*)d_in[10];
  const float* W3  = (const float*)d_in[11];
  const float* b3  = (const float*)d_in[12];
  const float* g3  = (const float*)d_in[13];
  const float* be3 = (const float*)d_in[14];
  const float* Wl  = (const float*)d_in[15];
  const float* bl  = (const float*)d_in[16];

  const int F = 64;
  const int N = in_sizes[0] / F;
  const int E = in_sizes[1] / 2;
  const int* row = ei;
  const int* col = ei + E;

  float* ws    = (float*)d_ws;
  float* dis   = ws;                              // N
  float* hbuf  = dis  + N;                        // N*32
  float* cbuf  = hbuf + (size_t)N * HID;          // N*32
  float* o1    = cbuf + (size_t)N * HID;          // N*32
  float* o2    = o1   + (size_t)N * HID;          // N*32
  float* o3    = o2   + (size_t)N * HID;          // N*32
  float* stats = o3   + (size_t)N * HID;          // 64

  const int T = 256;

  // symmetric normalization with self-loops: deg = 1 + sum_w, dis = deg^{-1/2}
  k_init_deg<<<(N + T - 1) / T, T, 0, stream>>>(dis, N);
  k_deg_edges<<<(E + T - 1) / T, T, 0, stream>>>(col, ew, dis, E);
  k_deg_rsqrt<<<(N + T - 1) / T, T, 0, stream>>>(dis, N);

  const int numT = (N + 15) / 16;
  const dim3 gemmB(32, 8);
  const dim3 gemmG((numT + 7) / 8, 2);
  const int elemB = (N * HID + T - 1) / T;
  const long long propT = (long long)(E + N) * 32;
  const int propB = (int)((propT + T - 1) / T);

  // ---- layer 1 ----
  k_gemm_wmma<64><<<gemmG, gemmB, 0, stream>>>(x, W1, hbuf, N);
  k_init_bias<<<elemB, T, 0, stream>>>(cbuf, b1, N * HID);
  k_prop<<<propB, T, 0, stream>>>(hbuf, row, col, ew, dis, cbuf, E, N);
  k_zero_stats<<<1, 64, 0, stream>>>(stats);
  k_bn_stats<<<512, T, 0, stream>>>(cbuf, stats, N);
  k_bn_apply<<<elemB, T, 0, stream>>>(cbuf, stats, g1, be1, o1, N);

  // ---- layer 2 ----
  k_gemm_wmma<32><<<gemmG, gemmB, 0, stream>>>(o1, W2, hbuf, N);
  k_init_bias<<<elemB, T, 0, stream>>>(cbuf, b2, N * HID);
  k_prop<<<propB, T, 0, stream>>>(hbuf, row, col, ew, dis, cbuf, E, N);
  k_zero_stats<<<1, 64, 0, stream>>>(stats);
  k_bn_stats<<<512, T, 0, stream>>>(cbuf, stats, N);
  k_bn_apply<<<elemB, T, 0, stream>>>(cbuf, stats, g2, be2, o2, N);

  // ---- layer 3 ----
  k_gemm_wmma<32><<<gemmG, gemmB, 0, stream>>>(o2, W3, hbuf, N);
  k_init_bias<<<elemB, T, 0, stream>>>(cbuf, b3, N * HID);
  k_prop<<<propB, T, 0, stream>>>(hbuf, row, col, ew, dis, cbuf, E, N);
  k_zero_stats<<<1, 64, 0, stream>>>(stats);
  k_bn_stats<<<512, T, 0, stream>>>(cbuf, stats, N);
  k_bn_apply<<<elemB, T, 0, stream>>>(cbuf, stats, g3, be3, o3, N);

  // ---- readout ----
  k_final<<<(N + T - 1) / T, T, 0, stream>>>(o1, o2, o3, Wl, bl, (float*)d_out, N);
}